// APRConv_5257039970541
// MI455X (gfx1250) — compile-verified
//
#include <hip/hip_runtime.h>
#include <hip/hip_bf16.h>

typedef __attribute__((ext_vector_type(2))) float v2f;
typedef __attribute__((ext_vector_type(8))) float v8f;

#define N_CIN   8
#define N_S     4
#define N_K2    9
#define KDIM    72          // CIN * K2
#define NCHUNK  18          // KDIM / 4
#define WAVES   8
#define BLOCK_T 256
#define GROW    34          // padded dword stride of one cj-pair row (16 pts * 2 + 2 pad)
#define GROWS   36          // KDIM / 2 pair-rows

__global__ __launch_bounds__(BLOCK_T) void apr_conv_wmma_kernel(
    const float* __restrict__ intens,   // (8, N)
    const float* __restrict__ weight,   // (8, 8, 4, 3, 3)
    const float* __restrict__ bias,     // (8,)
    const int*   __restrict__ nbr,      // (N, 9)
    const int*   __restrict__ sten,     // (N,)
    float*       __restrict__ out,      // (8, N)
    int N, int nTiles)
{
    // A-fragments with TWO stencil slices packed per 16-row A matrix:
    //   rows 0..7  = W[o][c][s = 2*sp]     (even stencil of the pair)
    //   rows 8..15 = W[o][c][s = 2*sp + 1] (odd  stencil of the pair)
    // index = kb*128 + sp*64 + lane*2 + v
    __shared__ float wfrag[NCHUNK * 2 * 64];             // 2304 f32 = 9 KB
    // Per-wave B staging: value for (cj, point p) at (cj>>1)*GROW + 2p + (cj&1)
    __shared__ float gbuf_all[WAVES][GROWS * GROW];      // ~4.9 KB per wave

    const int tid  = threadIdx.x;
    const int lane = tid & 31;
    const int wave = tid >> 5;
    const int p    = lane & 15;   // point within tile this lane serves
    const int h    = lane >> 4;   // half-wave: K-pair / channel range / stencil parity

    // ---- cooperative fill of weight fragments (once per block) ----
    for (int e = tid; e < NCHUNK * 2 * 64; e += BLOCK_T) {
        const int v  = e & 1;
        const int l  = (e >> 1) & 31;
        const int sp = (e >> 6) & 1;               // stencil pair: {0,1} or {2,3}
        const int kb = e >> 7;
        const int m  = l & 15;                     // A-matrix row
        const int k  = 4 * kb + 2 * (l >> 4) + v;  // A-matrix column (cj)
        const int o  = m & 7;                      // output channel
        const int s  = sp * 2 + (m >> 3);          // stencil slice for this row half
        const int c  = k / 9;
        const int j  = k - 9 * c;
        wfrag[e] = weight[((o * N_CIN + c) * N_S + s) * N_K2 + j];
    }
    __syncthreads();

    float bs[8];
    #pragma unroll
    for (int r = 0; r < 8; ++r) bs[r] = bias[r];

    float* gbuf = gbuf_all[wave];

    // loop-invariant byte-offset bases for the 4 channels this lane gathers
    unsigned cbase[4];
    #pragma unroll
    for (int ci = 0; ci < 4; ++ci)
        cbase[ci] = (unsigned)(h * 4 + ci) * (unsigned)N * 4u;

    const int waveGlobal = blockIdx.x * WAVES + wave;
    const int waveCount  = gridDim.x * WAVES;

    for (int t = waveGlobal; t < nTiles; t += waveCount) {
        const int base = t * 16;
        int pt = base + p;
        if (pt >= N) pt = N - 1;           // clamp tail; stores are guarded below

        // prefetch the next tile's index rows while this tile computes
        {
            int tn = t + waveCount;
            if (tn < nTiles) {
                const int ptn = tn * 16 + p;
                __builtin_prefetch(nbr + (size_t)ptn * N_K2, 0, 3);
            }
        }

        // ---- gather phase: lane fetches channels [4h, 4h+4) of point pt ----
        const int* ip = nbr + (size_t)pt * N_K2;
        int idx[9];
        #pragma unroll
        for (int j = 0; j < 9; ++j) idx[j] = ip[j];

        __builtin_amdgcn_wave_barrier();
        // Fused gather->LDS scatter (SADDR form): uniform 64-bit base in SGPRs,
        // per-lane 32-bit byte offset, per-lane LDS destination; ASYNCcnt.
        #pragma unroll
        for (int ci = 0; ci < 4; ++ci) {
            const int c = h * 4 + ci;
            #pragma unroll
            for (int j = 0; j < 9; ++j) {
                const int cj = c * 9 + j;
                const unsigned boff = ((unsigned)idx[j] << 2) + cbase[ci];
                const unsigned ldsb =
                    (unsigned)(uintptr_t)&gbuf[(cj >> 1) * GROW + 2 * p + (cj & 1)];
                asm volatile("global_load_async_to_lds_b32 %0, %1, %2"
                             :: "v"(ldsb), "v"(boff), "s"(intens) : "memory");
            }
        }
        asm volatile("s_wait_asynccnt 0x0" ::: "memory");
        __builtin_amdgcn_wave_barrier();

        // ---- 2 packed-stencil GEMM accumulations over K=72 (18 chunks of 4) ----
        v8f acc01 = {}, acc23 = {};
        #pragma unroll
        for (int kb = 0; kb < NCHUNK; ++kb) {
            const v2f b = *(const v2f*)&gbuf[(2 * kb + h) * GROW + 2 * p];
            const float* wf = &wfrag[kb * 128 + lane * 2];
            const v2f a01 = *(const v2f*)(wf + 0);
            const v2f a23 = *(const v2f*)(wf + 64);
            acc01 = __builtin_amdgcn_wmma_f32_16x16x4_f32(false, a01, false, b,
                                                          (short)0, acc01, false, false);
            acc23 = __builtin_amdgcn_wmma_f32_16x16x4_f32(false, a23, false, b,
                                                          (short)0, acc23, false, false);
        }
        __builtin_amdgcn_wave_barrier();

        // ---- per-point stencil select + bias + store ----
        // D layout: VGPR r, lanes 0-15 hold (M=r, col=lane)      -> even stencil
        //                  lanes 16-31 hold (M=8+r, col=lane-16) -> odd  stencil
        // So lane (h*16 + p) holds the stencil-(2k+h) result for point p:
        // exactly one half-wave matches each point's stencil parity.
        const int s_id = sten[pt];
        if (((s_id & 1) == h) && (base + p) < N) {
            #pragma unroll
            for (int r = 0; r < 8; ++r) {
                const float v = (s_id < 2) ? acc01[r] : acc23[r];
                out[(size_t)r * N + base + p] = v + bs[r];
            }
        }
    }
}

extern "C" void kernel_launch(void* const* d_in, const int* in_sizes, int n_in,
                              void* d_out, int out_size, void* d_ws, size_t ws_size,
                              hipStream_t stream) {
    const float* intens = (const float*)d_in[0];   // (1, 8, N) f32
    const float* weight = (const float*)d_in[1];   // (8, 8, 4, 3, 3) f32
    const float* bias   = (const float*)d_in[2];   // (8,) f32
    const int*   nbr    = (const int*)d_in[3];     // (N, 9) i32
    const int*   sten   = (const int*)d_in[4];     // (N,) i32
    float* out = (float*)d_out;                    // (1, 8, N) f32

    const int N      = in_sizes[0] / N_CIN;
    const int nTiles = (N + 15) / 16;

    const int blocks = 512;                        // grid-stride over 125k tiles
    apr_conv_wmma_kernel<<<blocks, BLOCK_T, 0, stream>>>(
        intens, weight, bias, nbr, sten, out, N, nTiles);
}